// GraphSAGE_59356448031328
// MI455X (gfx1250) — compile-verified
//
#include <hip/hip_runtime.h>

#define N_NODES   50000
#define N_EDGES   800000
#define HID       128
#define N_CLASSES 2
#define N_GRAPHS  500
#define N_LAYERS  7

typedef __attribute__((ext_vector_type(2))) float v2f;
typedef __attribute__((ext_vector_type(8))) float v8f;

// ---------------------------------------------------------------- utilities
__global__ void fill_zero(float* __restrict__ p, int n) {
    int i = blockIdx.x * blockDim.x + threadIdx.x;
    int stride = gridDim.x * blockDim.x;
    for (; i < n; i += stride) p[i] = 0.0f;
}

// ---------------------------------------------------------------- degree
__global__ void deg_count(const int* __restrict__ dst, float* __restrict__ deg) {
    int e = blockIdx.x * blockDim.x + threadIdx.x;
    if (e < N_EDGES) unsafeAtomicAdd(&deg[dst[e]], 1.0f);
}

__global__ void deg_invert(float* __restrict__ deg) {
    int n = blockIdx.x * blockDim.x + threadIdx.x;
    if (n < N_NODES) {
        float d = deg[n];
        deg[n] = 1.0f / fmaxf(d, 1.0f);
    }
}

// ---------------------------------------------------------------- edge scatter (mean-agg numerator)
// one wave per edge: lane loads float4 of h[src], atomically adds into agg[dst]
__global__ __launch_bounds__(256) void edge_scatter(
    const float* __restrict__ h, const int* __restrict__ src,
    const int* __restrict__ dst, float* __restrict__ agg) {
    int e = blockIdx.x * 8 + (threadIdx.x >> 5);
    int lane = threadIdx.x & 31;
    if (e >= N_EDGES) return;
    int s = src[e];
    int d = dst[e];
    const float4 v = *((const float4*)(h + (size_t)s * HID) + lane);
    float* ap = agg + (size_t)d * HID + lane * 4;
    unsafeAtomicAdd(ap + 0, v.x);
    unsafeAtomicAdd(ap + 1, v.y);
    unsafeAtomicAdd(ap + 2, v.z);
    unsafeAtomicAdd(ap + 3, v.w);
}

// ---------------------------------------------------------------- fused SAGE layer GEMM (fp32 WMMA)
// h_out = relu_opt( (agg * inv_deg) @ Wl + bl + h @ Wr )
// one wave computes one 16x16 output tile; block = 8 waves = all 8 N-tiles of one M-tile
__global__ __launch_bounds__(256) void sage_gemm(
    const float* __restrict__ agg, const float* __restrict__ h,
    const float* __restrict__ inv_deg,
    const float* __restrict__ Wl, const float* __restrict__ Wr,
    const float* __restrict__ bl, float* __restrict__ hout, int relu) {
    const int mtile = blockIdx.x;             // 0..3124
    const int ntile = threadIdx.x >> 5;       // 0..7
    const int lane  = threadIdx.x & 31;
    const int half  = lane >> 4;              // K-half selector for A/B fragments
    const int l15   = lane & 15;
    const int m = mtile * 16 + l15;           // A row handled by this lane
    const int n = ntile * 16 + l15;           // B/C/D column handled by this lane

    const float s = inv_deg[m];               // fold mean-normalization into A load
    const float bias = bl[n];
    v8f c;
#pragma unroll
    for (int r = 0; r < 8; ++r) c[r] = bias;  // C init = bias broadcast over rows

    const float* arow = agg + (size_t)m * HID + half * 2;  // contiguous K-pair per lane
    const float* hrow = h   + (size_t)m * HID + half * 2;

#pragma unroll 4
    for (int kb = 0; kb < HID; kb += 4) {
        v2f a = *(const v2f*)(arow + kb);     // A frag: K = kb+2*half, kb+2*half+1
        a.x *= s; a.y *= s;
        const int k0 = kb + half * 2;
        v2f b;
        b.x = Wl[(k0    ) * HID + n];         // B frag: rows k0, k0+1; column n
        b.y = Wl[(k0 + 1) * HID + n];
        c = __builtin_amdgcn_wmma_f32_16x16x4_f32(false, a, false, b,
                                                  (short)0, c, false, false);
    }
#pragma unroll 4
    for (int kb = 0; kb < HID; kb += 4) {
        v2f a = *(const v2f*)(hrow + kb);
        const int k0 = kb + half * 2;
        v2f b;
        b.x = Wr[(k0    ) * HID + n];
        b.y = Wr[(k0 + 1) * HID + n];
        c = __builtin_amdgcn_wmma_f32_16x16x4_f32(false, a, false, b,
                                                  (short)0, c, false, false);
    }

    // D layout: VGPR r -> row mtile*16 + half*8 + r, column n
#pragma unroll
    for (int r = 0; r < 8; ++r) {
        float v = c[r];
        if (relu) v = fmaxf(v, 0.0f);
        hout[(size_t)(mtile * 16 + half * 8 + r) * HID + n] = v;
    }
}

// ---------------------------------------------------------------- global add pool
__global__ __launch_bounds__(256) void pool_kernel(
    const float* __restrict__ h, const int* __restrict__ batch,
    float* __restrict__ pooled) {
    int node = blockIdx.x * 2 + (threadIdx.x >> 7);
    int j = threadIdx.x & 127;
    if (node >= N_NODES) return;
    int g = batch[node];
    unsafeAtomicAdd(&pooled[(size_t)g * HID + j], h[(size_t)node * HID + j]);
}

// ---------------------------------------------------------------- classifier head (tiny, scalar)
__global__ void out_kernel(const float* __restrict__ pooled,
                           const float* __restrict__ Wout,
                           const float* __restrict__ bout,
                           float* __restrict__ out) {
    int t = blockIdx.x * blockDim.x + threadIdx.x;
    if (t >= N_GRAPHS * N_CLASSES) return;
    int g = t >> 1, cidx = t & 1;
    float acc = bout[cidx];
    const float* pr = pooled + (size_t)g * HID;
#pragma unroll 8
    for (int k = 0; k < HID; ++k) acc += pr[k] * Wout[k * N_CLASSES + cidx];
    out[t] = acc;
}

// ---------------------------------------------------------------- driver
extern "C" void kernel_launch(void* const* d_in, const int* in_sizes, int n_in,
                              void* d_out, int out_size, void* d_ws, size_t ws_size,
                              hipStream_t stream) {
    const float* x    = (const float*)d_in[0];
    const int*   ei   = (const int*)d_in[1];
    const int*   batch= (const int*)d_in[2];
    const float* Wl   = (const float*)d_in[3];
    const float* Wr   = (const float*)d_in[4];
    const float* bl   = (const float*)d_in[5];
    const float* Wout = (const float*)d_in[6];
    const float* bout = (const float*)d_in[7];
    const int* src = ei;             // edge_index[0]
    const int* dst = ei + N_EDGES;   // edge_index[1]

    float* ws = (float*)d_ws;
    size_t o = 0;
    float* inv_deg = ws + o; o += ((N_NODES + 255) / 256) * 256;
    float* agg     = ws + o; o += (size_t)N_NODES * HID;
    float* h0      = ws + o; o += (size_t)N_NODES * HID;
    float* h1      = ws + o; o += (size_t)N_NODES * HID;
    float* pooled  = ws + o; o += (size_t)N_GRAPHS * HID;

    // degree -> inv_deg (fresh every call: deterministic)
    fill_zero<<<256, 256, 0, stream>>>(inv_deg, N_NODES);
    deg_count<<<(N_EDGES + 255) / 256, 256, 0, stream>>>(dst, inv_deg);
    deg_invert<<<(N_NODES + 255) / 256, 256, 0, stream>>>(inv_deg);

    const float* hcur = x;
    float* bufs[2] = {h0, h1};
    for (int i = 0; i < N_LAYERS; ++i) {
        fill_zero<<<2048, 256, 0, stream>>>(agg, N_NODES * HID);
        edge_scatter<<<(N_EDGES + 7) / 8, 256, 0, stream>>>(hcur, src, dst, agg);
        float* hnext = bufs[i & 1];
        sage_gemm<<<N_NODES / 16, 256, 0, stream>>>(
            agg, hcur, inv_deg,
            Wl + (size_t)i * HID * HID, Wr + (size_t)i * HID * HID,
            bl + (size_t)i * HID, hnext, (i < N_LAYERS - 1) ? 1 : 0);
        hcur = hnext;
    }

    fill_zero<<<64, 256, 0, stream>>>(pooled, N_GRAPHS * HID);
    pool_kernel<<<N_NODES / 2, 256, 0, stream>>>(hcur, batch, pooled);
    out_kernel<<<(N_GRAPHS * N_CLASSES + 255) / 256, 256, 0, stream>>>(
        pooled, Wout, bout, (float*)d_out);
}